// DocumentEncoder_1881195675853
// MI455X (gfx1250) — compile-verified
//
#include <hip/hip_runtime.h>
#include <cstddef>

#define BDOCS 256
#define LTOK  512
#define DDIM  1024

typedef __attribute__((ext_vector_type(2))) float v2f;
typedef __attribute__((ext_vector_type(8))) float v8f;

// ---------------------------------------------------------------------------
// Kernel 1: hidden[b,:] = (1/L) * sum_l emb_table[encodes[b,l], :]
// One workgroup per doc; thread t owns columns 4t..4t+3 (coalesced float4).
// ---------------------------------------------------------------------------
__global__ __launch_bounds__(256) void hidden_mean_kernel(
    const int* __restrict__ encodes, const float* __restrict__ emb,
    float* __restrict__ hidden) {
  const int b = blockIdx.x;
  const int t = threadIdx.x;
  __shared__ int idx[LTOK];
  for (int l = t; l < LTOK; l += 256) idx[l] = encodes[b * LTOK + l];
  __syncthreads();

  const int c = t * 4;
  float4 acc = make_float4(0.f, 0.f, 0.f, 0.f);
  for (int l = 0; l < LTOK; ++l) {
    if (l + 8 < LTOK)
      __builtin_prefetch(emb + (size_t)idx[l + 8] * DDIM + c, 0, 1);
    const float4 e = *(const float4*)(emb + (size_t)idx[l] * DDIM + c);
    acc.x += e.x; acc.y += e.y; acc.z += e.z; acc.w += e.w;
  }
  const float s = 1.0f / (float)LTOK;
  float4 r = make_float4(acc.x * s, acc.y * s, acc.z * s, acc.w * s);
  *(float4*)(hidden + (size_t)b * DDIM + c) = r;
}

// ---------------------------------------------------------------------------
// Kernel 2: V = hidden @ W_b^T  (M=256, N=1024, K=1024), fp32 WMMA.
// V[b,d] = sum_k hidden[b,k] * W_b[d,k]  -> B-fragment reads W_b rows directly.
// One wave per 16x16 output tile; K stepped by 4 with V_WMMA_F32_16X16X4_F32.
// A 16x4 layout: lanes 0-15 hold K=0,1 ; lanes 16-31 hold K=2,3 (float2 loads).
// B 4x16 layout mirrors A with N across lanes.
// ---------------------------------------------------------------------------
__global__ __launch_bounds__(256) void bilinear_wmma_kernel(
    const float* __restrict__ hidden, const float* __restrict__ Wb,
    float* __restrict__ V) {
  const int lane = threadIdx.x & 31;
  const int wave = blockIdx.x * 8 + (threadIdx.x >> 5);  // 128 blocks * 8 waves = 1024 tiles
  const int tn = wave & 63;   // 64 N-tiles
  const int tm = wave >> 6;   // 16 M-tiles
  const int half = lane >> 4; // 0: K=0,1   1: K=2,3
  const int lid  = lane & 15;

  const float* arow = hidden + (size_t)(tm * 16 + lid) * DDIM + half * 2;
  const float* brow = Wb     + (size_t)(tn * 16 + lid) * DDIM + half * 2;

  v8f acc = {};
#pragma unroll 4
  for (int k = 0; k < DDIM; k += 4) {
    v2f a = *(const v2f*)(arow + k);
    v2f bfrag = *(const v2f*)(brow + k);
    acc = __builtin_amdgcn_wmma_f32_16x16x4_f32(false, a, false, bfrag,
                                                (short)0, acc, false, false);
  }

  // D layout: VGPR r -> lanes 0-15: M=r, lanes 16-31: M=8+r; N = lane&15.
  const int n = tn * 16 + lid;
  const int mbase = tm * 16 + half * 8;
#pragma unroll
  for (int r = 0; r < 8; ++r)
    V[(size_t)(mbase + r) * DDIM + n] = acc[r];
}

// ---------------------------------------------------------------------------
// Kernel 3: per-doc scores = emb . v, softmax over L, ct = sum_l attn*emb.
// One workgroup (8 waves) per doc. v staged in LDS (4KB); scores in LDS (2KB).
// ---------------------------------------------------------------------------
__global__ __launch_bounds__(256) void attn_context_kernel(
    const int* __restrict__ encodes, const float* __restrict__ emb,
    const float* __restrict__ V, float* __restrict__ ct) {
  const int b = blockIdx.x;
  const int t = threadIdx.x;
  const int wave = t >> 5;
  const int lane = t & 31;

  __shared__ int   idx[LTOK];
  __shared__ float vsh[DDIM];
  __shared__ float sc[LTOK];
  __shared__ float red[8];

  for (int l = t; l < LTOK; l += 256) idx[l] = encodes[b * LTOK + l];
  for (int d0 = t; d0 < DDIM; d0 += 256) vsh[d0] = V[(size_t)b * DDIM + d0];
  __syncthreads();

  // ---- scores: one token per wave, dot over D with LDS-resident v ----
  for (int l = wave; l < LTOK; l += 8) {
    const float* row = emb + (size_t)idx[l] * DDIM;
    float s = 0.f;
#pragma unroll
    for (int i = 0; i < DDIM; i += 128) {
      const float4 e  = *(const float4*)(row + i + lane * 4);
      const float4 vv = *(const float4*)(vsh + i + lane * 4);
      s += e.x * vv.x + e.y * vv.y + e.z * vv.z + e.w * vv.w;
    }
#pragma unroll
    for (int off = 16; off > 0; off >>= 1) s += __shfl_xor(s, off, 32);
    if (lane == 0) sc[l] = s;
  }
  __syncthreads();

  // ---- softmax over L (block-wide, wave32 reductions) ----
  float mx = -3.402823466e38f;
  for (int l = t; l < LTOK; l += 256) mx = fmaxf(mx, sc[l]);
#pragma unroll
  for (int off = 16; off > 0; off >>= 1) mx = fmaxf(mx, __shfl_xor(mx, off, 32));
  if (lane == 0) red[wave] = mx;
  __syncthreads();
  mx = red[0];
#pragma unroll
  for (int i = 1; i < 8; ++i) mx = fmaxf(mx, red[i]);
  __syncthreads();

  float sum = 0.f;
  for (int l = t; l < LTOK; l += 256) {
    float e = __expf(sc[l] - mx);
    sc[l] = e;
    sum += e;
  }
#pragma unroll
  for (int off = 16; off > 0; off >>= 1) sum += __shfl_xor(sum, off, 32);
  if (lane == 0) red[wave] = sum;
  __syncthreads();
  float tot = 0.f;
#pragma unroll
  for (int i = 0; i < 8; ++i) tot += red[i];
  const float inv = 1.0f / tot;

  // ---- context: ct[b, 4t..4t+3] = sum_l w[l] * emb_row[4t..4t+3] ----
  const int c = t * 4;
  float4 acc = make_float4(0.f, 0.f, 0.f, 0.f);
  for (int l = 0; l < LTOK; ++l) {
    if (l + 8 < LTOK)
      __builtin_prefetch(emb + (size_t)idx[l + 8] * DDIM + c, 0, 1);
    const float4 e = *(const float4*)(emb + (size_t)idx[l] * DDIM + c);
    const float w = sc[l] * inv;
    acc.x += w * e.x; acc.y += w * e.y; acc.z += w * e.z; acc.w += w * e.w;
  }
  *(float4*)(ct + (size_t)b * DDIM + c) = acc;
}

// ---------------------------------------------------------------------------
extern "C" void kernel_launch(void* const* d_in, const int* in_sizes, int n_in,
                              void* d_out, int out_size, void* d_ws, size_t ws_size,
                              hipStream_t stream) {
  (void)in_sizes; (void)n_in; (void)out_size; (void)ws_size;
  const int*   encodes = (const int*)d_in[0];
  const float* emb     = (const float*)d_in[1];
  const float* Wb      = (const float*)d_in[2];
  float* out = (float*)d_out;

  float* hidden = (float*)d_ws;                       // [B, D] = 1 MB
  float* V      = hidden + (size_t)BDOCS * DDIM;      // [B, D] = 1 MB

  hidden_mean_kernel<<<BDOCS, 256, 0, stream>>>(encodes, emb, hidden);
  bilinear_wmma_kernel<<<128, 256, 0, stream>>>(hidden, Wb, V);
  attn_context_kernel<<<BDOCS, 256, 0, stream>>>(encodes, emb, V, out);
}